// TransformerConvNet_41987600285942
// MI455X (gfx1250) — compile-verified
//
#include <hip/hip_runtime.h>
#include <hip/hip_bf16.h>
#include <math.h>

// ---------------------------------------------------------------------------
// Problem constants (from reference): B=8, N=128, EMB_H=128, HEADS=8, DH=64.
// ---------------------------------------------------------------------------
#define BB 8
#define NN 128
#define EMB_H 128
#define HEADS 8
#define DHD 64
#define CCH (HEADS * DHD)   // 512

typedef float v2f __attribute__((ext_vector_type(2)));
typedef float v8f __attribute__((ext_vector_type(8)));

// D = A(16x4 f32) x B(4x16 f32) + C(16x16 f32), wave32 WMMA.
__device__ __forceinline__ v8f wmma4(v2f a, v2f b, v8f c) {
    return __builtin_amdgcn_wmma_f32_16x16x4_f32(
        /*neg_a=*/false, a, /*neg_b=*/false, b,
        /*c_mod=*/(short)0, c, /*reuse_a=*/false, /*reuse_b=*/false);
}

// ---------------------------------------------------------------------------
// k_prep: Et[b,i,j] = adj[b,j,i] if 0<adj<0.5 else 0 ; Atf likewise 1/0.
// ---------------------------------------------------------------------------
__global__ void k_prep(const float* __restrict__ adj,
                       float* __restrict__ Et, float* __restrict__ Atf) {
    int idx = blockIdx.x * blockDim.x + threadIdx.x;   // over B*N*N = 131072
    if (idx >= BB * NN * NN) return;
    int b = idx >> 14;
    int rem = idx & 16383;
    int i = rem >> 7, j = rem & 127;
    float a = adj[(size_t)b * NN * NN + (size_t)j * NN + i];
    bool m = (a > 0.f) && (a < 0.5f);
    Et[idx]  = m ? a : 0.f;
    Atf[idx] = m ? 1.f : 0.f;
}

// ---------------------------------------------------------------------------
// k_hnode: h_node[row, c] = [feat(11) | entity_embed[etype](8)] @ W1[:19] + b1
// ---------------------------------------------------------------------------
__global__ __launch_bounds__(128) void k_hnode(
    const float* __restrict__ node_obs, const float* __restrict__ emb,
    const float* __restrict__ W1, const float* __restrict__ b1,
    float* __restrict__ hnode) {
    int row = blockIdx.x;        // 0..1023  (b*N + j)
    int c = threadIdx.x;         // 0..127
    const float* obs = node_obs + (size_t)row * 12;
    int et = (int)obs[11];
    float acc = b1[c];
    for (int t = 0; t < 11; ++t) acc += obs[t] * W1[t * EMB_H + c];
    for (int t = 0; t < 8;  ++t) acc += emb[et * 8 + t] * W1[(11 + t) * EMB_H + c];
    hnode[(size_t)row * EMB_H + c] = acc;
}

// ---------------------------------------------------------------------------
// k_emb: per (b,i) block. For j in 0..127 (two halves of 64):
//   h1 = LN(relu(h_node[b,j] + Et[b,i,j]*w_edge))        (LDS, 64x128)
//   h2 = h1 @ Wh + bh                                    (WMMA GEMM 64x128x128)
//   h3 = LN(relu(h2))
//   x0[b,i,:] += Atf[b,i,j] * h3
// ---------------------------------------------------------------------------
__global__ __launch_bounds__(256) void k_emb(
    const float* __restrict__ hnode, const float* __restrict__ Et,
    const float* __restrict__ Atf, const float* __restrict__ W1,
    const float* __restrict__ g1, const float* __restrict__ b1n,
    const float* __restrict__ Wh, const float* __restrict__ bh,
    const float* __restrict__ g2, const float* __restrict__ b2n,
    float* __restrict__ x0) {
    const int blk = blockIdx.x;                    // b*N + i
    const int b = blk >> 7;
    const int i = blk & 127;
    const int tid = threadIdx.x;
    const int wv = tid >> 5;
    const int lane = tid & 31;
    const int lo = lane & 15, hi = lane >> 4;

    __shared__ float sH[64 * EMB_H];
    __shared__ float e_sh[64], a_sh[64];
    __shared__ float xacc[EMB_H];
    const float* w_edge = W1 + 19 * EMB_H;

    if (tid < EMB_H) xacc[tid] = 0.f;

    for (int half = 0; half < 2; ++half) {
        __syncthreads();
        if (tid < 64) {
            int j = half * 64 + tid;
            e_sh[tid] = Et[(size_t)b * NN * NN + (size_t)i * NN + j];
            a_sh[tid] = Atf[(size_t)b * NN * NN + (size_t)i * NN + j];
        }
        __syncthreads();

        // pass 1: build h1 rows (8 rows per wave, lane covers c, c+32, c+64, c+96)
        for (int rr = 0; rr < 8; ++rr) {
            int jl = wv * 8 + rr;
            int j = half * 64 + jl;
            float e = e_sh[jl];
            float t4[4]; float s = 0.f, ss = 0.f;
            for (int cc = 0; cc < 4; ++cc) {
                int c = lane + 32 * cc;
                float t = hnode[((size_t)b * NN + j) * EMB_H + c] + e * w_edge[c];
                t = fmaxf(t, 0.f);
                t4[cc] = t; s += t; ss += t * t;
            }
            for (int o = 16; o > 0; o >>= 1) {
                s += __shfl_xor(s, o, 32); ss += __shfl_xor(ss, o, 32);
            }
            float mu = s * (1.f / 128.f);
            float var = ss * (1.f / 128.f) - mu * mu;
            float inv = rsqrtf(var + 1e-5f);
            for (int cc = 0; cc < 4; ++cc) {
                int c = lane + 32 * cc;
                sH[jl * EMB_H + c] = (t4[cc] - mu) * inv * g1[c] + b1n[c];
            }
        }
        __syncthreads();

        // GEMM: h2(64x128) = h1(64x128) @ Wh(128x128)
        // wave tiles: rows 16*(wv&3), cols 64*(wv>>2) .. +64 (4 tiles of 16x16)
        int m0 = 16 * (wv & 3);
        int n0 = 64 * (wv >> 2);
        v8f zero = {0.f, 0.f, 0.f, 0.f, 0.f, 0.f, 0.f, 0.f};
        v8f acc[4] = {zero, zero, zero, zero};
        for (int kk = 0; kk < EMB_H; kk += 4) {
            v2f a;
            a.x = sH[(m0 + lo) * EMB_H + kk + 2 * hi + 0];
            a.y = sH[(m0 + lo) * EMB_H + kk + 2 * hi + 1];
            for (int t = 0; t < 4; ++t) {
                int n = n0 + 16 * t + lo;
                v2f bf;
                bf.x = Wh[(size_t)(kk + 2 * hi + 0) * EMB_H + n];
                bf.y = Wh[(size_t)(kk + 2 * hi + 1) * EMB_H + n];
                acc[t] = wmma4(a, bf, acc[t]);
            }
        }
        __syncthreads();
        // write h2 + bh back into sH
        for (int t = 0; t < 4; ++t) {
            int n = n0 + 16 * t + lo;
            float bb = bh[n];
            for (int vv = 0; vv < 8; ++vv) {
                int m = m0 + vv + 8 * hi;
                sH[m * EMB_H + n] = acc[t][vv] + bb;
            }
        }
        __syncthreads();

        // pass 2: relu + LN + masked accumulate
        for (int rr = 0; rr < 8; ++rr) {
            int jl = wv * 8 + rr;
            float t4[4]; float s = 0.f, ss = 0.f;
            for (int cc = 0; cc < 4; ++cc) {
                int c = lane + 32 * cc;
                float u = fmaxf(sH[jl * EMB_H + c], 0.f);
                t4[cc] = u; s += u; ss += u * u;
            }
            for (int o = 16; o > 0; o >>= 1) {
                s += __shfl_xor(s, o, 32); ss += __shfl_xor(ss, o, 32);
            }
            float mu = s * (1.f / 128.f);
            float var = ss * (1.f / 128.f) - mu * mu;
            float inv = rsqrtf(var + 1e-5f);
            float aj = a_sh[jl];
            for (int cc = 0; cc < 4; ++cc) {
                int c = lane + 32 * cc;
                float h3 = (t4[cc] - mu) * inv * g2[c] + b2n[c];
                atomicAdd(&xacc[c], aj * h3);
            }
        }
    }
    __syncthreads();
    if (tid < EMB_H) x0[(size_t)blk * EMB_H + tid] = xacc[tid];
}

// ---------------------------------------------------------------------------
// k_gemm_bias: Y(1024 x 512) = X(1024 x K) @ W(K x 512) + bias. WMMA tiles.
// grid (8,8): block covers 128 rows x 64 cols; 8 waves * 4 tiles.
// ---------------------------------------------------------------------------
__global__ __launch_bounds__(256) void k_gemm_bias(
    const float* __restrict__ X, int lda,
    const float* __restrict__ W, const float* __restrict__ bias,
    float* __restrict__ Y, int K) {
    const int tid = threadIdx.x;
    const int wv = tid >> 5, lane = tid & 31, lo = lane & 15, hi = lane >> 4;
    const int m0 = blockIdx.x * 128 + wv * 16;
    const int n0 = blockIdx.y * 64;
    v8f zero = {0.f, 0.f, 0.f, 0.f, 0.f, 0.f, 0.f, 0.f};
    v8f acc[4] = {zero, zero, zero, zero};
    for (int kk = 0; kk < K; kk += 4) {
        if (kk + 8 < K)  // speculative prefetch of upcoming W rows (global_prefetch_b8)
            __builtin_prefetch(&W[(size_t)(kk + 8 + 2 * hi) * CCH + n0 + lo], 0, 0);
        v2f a;
        const float* xr = X + (size_t)(m0 + lo) * lda + kk + 2 * hi;
        a.x = xr[0]; a.y = xr[1];
        for (int t = 0; t < 4; ++t) {
            int n = n0 + 16 * t + lo;
            v2f bf;
            bf.x = W[(size_t)(kk + 2 * hi + 0) * CCH + n];
            bf.y = W[(size_t)(kk + 2 * hi + 1) * CCH + n];
            acc[t] = wmma4(a, bf, acc[t]);
        }
    }
    for (int t = 0; t < 4; ++t) {
        int n = n0 + 16 * t + lo;
        float bb = bias[n];
        for (int vv = 0; vv < 8; ++vv) {
            int m = m0 + vv + 8 * hi;
            Y[(size_t)m * CCH + n] = acc[t][vv] + bb;
        }
    }
}

// ---------------------------------------------------------------------------
// k_attn: block = (i-chunk ci, head h, batch b). 32 rows of i per block.
//   S = Q(32x64) @ K^T            (WMMA)
//   scores = (S + Et*qe)/8, masked softmax over j, ae = sum alpha*Et
//   O = alpha(32x128) @ V(128x64) (WMMA)
//   xout = relu(O + ae*we + skip)
// ---------------------------------------------------------------------------
#define SST 132
__global__ __launch_bounds__(256) void k_attn(
    const float* __restrict__ q, const float* __restrict__ k,
    const float* __restrict__ v, const float* __restrict__ skip,
    const float* __restrict__ Et, const float* __restrict__ Atf,
    const float* __restrict__ We, float* __restrict__ xout) {
    const int ci = blockIdx.x;   // 0..3
    const int h  = blockIdx.y;   // 0..7
    const int b  = blockIdx.z;   // 0..7
    const int i0 = ci * 32;
    const int tid = threadIdx.x, wv = tid >> 5, lane = tid & 31;
    const int lo = lane & 15, hi = lane >> 4;

    __shared__ float sS[32 * SST];
    __shared__ float qe_sh[32], ae_sh[32], we_sh[DHD];

    if (tid < DHD) we_sh[tid] = We[h * DHD + tid];

    // qe[i] = q[b,i,h,:] . we[h,:]   (wave wv owns rows 4wv..4wv+3)
    for (int rr = 0; rr < 4; ++rr) {
        int r = wv * 4 + rr;
        int i = i0 + r;
        const float* qrow = q + ((size_t)(b * NN + i)) * CCH + h * DHD;
        float p = qrow[lane] * We[h * DHD + lane] +
                  qrow[lane + 32] * We[h * DHD + lane + 32];
        for (int o = 16; o > 0; o >>= 1) p += __shfl_xor(p, o, 32);
        if (lane == 0) qe_sh[r] = p;
    }

    // GEMM1: S(32x128) = Q @ K^T; wave wv -> tiles (mt = wv>>2, nt = 2*(wv&3)+{0,1})
    {
        int mt = wv >> 2;
        int ntb = 2 * (wv & 3);
        v8f zero = {0.f, 0.f, 0.f, 0.f, 0.f, 0.f, 0.f, 0.f};
        v8f acc[2] = {zero, zero};
        for (int kk = 0; kk < DHD; kk += 4) {
            v2f a;
            const float* qrow = q + ((size_t)(b * NN + i0 + mt * 16 + lo)) * CCH
                              + h * DHD + kk + 2 * hi;
            a.x = qrow[0]; a.y = qrow[1];
            for (int t = 0; t < 2; ++t) {
                int j = (ntb + t) * 16 + lo;
                const float* krow = k + ((size_t)(b * NN + j)) * CCH
                                  + h * DHD + kk + 2 * hi;
                v2f bf; bf.x = krow[0]; bf.y = krow[1];
                acc[t] = wmma4(a, bf, acc[t]);
            }
        }
        for (int t = 0; t < 2; ++t) {
            int n = (ntb + t) * 16 + lo;
            for (int vv = 0; vv < 8; ++vv) {
                int m = mt * 16 + vv + 8 * hi;
                sS[m * SST + n] = acc[t][vv];
            }
        }
    }
    __syncthreads();

    // masked softmax over j (rows r = 4wv..4wv+3; lane covers j, j+32, j+64, j+96)
    for (int rr = 0; rr < 4; ++rr) {
        int r = wv * 4 + rr;
        int i = i0 + r;
        float qe = qe_sh[r];
        const float* Erow = Et + (size_t)(b * NN + i) * NN;
        const float* Arow = Atf + (size_t)(b * NN + i) * NN;
        float sv[4], ev[4], msk[4];
        float mrow = -3.4028235e38f;
        for (int cc = 0; cc < 4; ++cc) {
            int j = lane + 32 * cc;
            float s = (sS[r * SST + j] + Erow[j] * qe) * 0.125f;
            float mk = Arow[j];
            msk[cc] = mk;
            sv[cc] = (mk > 0.f) ? s : -3.4028235e38f;
            mrow = fmaxf(mrow, sv[cc]);
        }
        for (int o = 16; o > 0; o >>= 1) mrow = fmaxf(mrow, __shfl_xor(mrow, o, 32));
        float ssum = 0.f;
        for (int cc = 0; cc < 4; ++cc) {
            float e = (msk[cc] > 0.f) ? __expf(sv[cc] - mrow) : 0.f;
            ev[cc] = e; ssum += e;
        }
        for (int o = 16; o > 0; o >>= 1) ssum += __shfl_xor(ssum, o, 32);
        float inv = 1.f / ((ssum <= 0.f) ? 1.f : ssum);
        float aesum = 0.f;
        for (int cc = 0; cc < 4; ++cc) {
            int j = lane + 32 * cc;
            float al = ev[cc] * inv;
            sS[r * SST + j] = al;
            aesum += al * Erow[j];
        }
        for (int o = 16; o > 0; o >>= 1) aesum += __shfl_xor(aesum, o, 32);
        if (lane == 0) ae_sh[r] = aesum;
    }
    __syncthreads();

    // GEMM2: O(32x64) = alpha(32x128) @ V(128x64); one 16x16 tile per wave
    {
        int mt = wv >> 2, nt = wv & 3;
        v8f acc = {0.f, 0.f, 0.f, 0.f, 0.f, 0.f, 0.f, 0.f};
        for (int kk = 0; kk < NN; kk += 4) {
            v2f a;
            a.x = sS[(mt * 16 + lo) * SST + kk + 2 * hi + 0];
            a.y = sS[(mt * 16 + lo) * SST + kk + 2 * hi + 1];
            v2f bf;
            bf.x = v[((size_t)(b * NN + kk + 2 * hi + 0)) * CCH + h * DHD + nt * 16 + lo];
            bf.y = v[((size_t)(b * NN + kk + 2 * hi + 1)) * CCH + h * DHD + nt * 16 + lo];
            acc = wmma4(a, bf, acc);
        }
        int n = nt * 16 + lo;
        for (int vv = 0; vv < 8; ++vv) {
            int m = mt * 16 + vv + 8 * hi;
            int i = i0 + m;
            float val = acc[vv] + ae_sh[m] * we_sh[n]
                      + skip[(size_t)(b * NN + i) * CCH + h * DHD + n];
            xout[(size_t)(b * NN + i) * CCH + h * DHD + n] = fmaxf(val, 0.f);
        }
    }
}

// ---------------------------------------------------------------------------
// k_gather: out[b, :] = x[b, agent_id[b], :]
// ---------------------------------------------------------------------------
__global__ void k_gather(const float* __restrict__ x, const int* __restrict__ agent,
                         float* __restrict__ out) {
    int t = blockIdx.x * blockDim.x + threadIdx.x;
    if (t >= BB * CCH) return;
    int b = t >> 9, c = t & 511;
    int i = agent[b];
    out[t] = x[(size_t)(b * NN + i) * CCH + c];
}

// ---------------------------------------------------------------------------
// Host side
// ---------------------------------------------------------------------------
extern "C" void kernel_launch(void* const* d_in, const int* in_sizes, int n_in,
                              void* d_out, int out_size, void* d_ws, size_t ws_size,
                              hipStream_t stream) {
    (void)in_sizes; (void)n_in; (void)out_size; (void)ws_size;
    const float* node_obs     = (const float*)d_in[0];   // (8,128,12)
    const float* adj          = (const float*)d_in[1];   // (8,128,128)
    const int*   agent        = (const int*)d_in[2];     // (8,)
    const float* entity_embed = (const float*)d_in[3];   // (4,8)
    const float* emb_W1       = (const float*)d_in[4];   // (20,128)
    const float* emb_b1       = (const float*)d_in[5];   // (128,)
    const float* ln1_g        = (const float*)d_in[6];
    const float* ln1_b        = (const float*)d_in[7];
    const float* emb_Wh       = (const float*)d_in[8];   // (128,128)
    const float* emb_bh       = (const float*)d_in[9];
    const float* lnh_g        = (const float*)d_in[10];
    const float* lnh_b        = (const float*)d_in[11];
    // gnn layers at 12 + 9*l : Wq,bq,Wk,bk,Wv,bv,We,Wskip,bskip

    float* ws    = (float*)d_ws;
    float* Et    = ws;                 // 131072
    float* Atf   = ws + 131072;        // 131072
    float* hnode = ws + 262144;        // 131072
    float* x0    = ws + 393216;        // 131072
    float* q     = ws + 524288;        // 524288
    float* kk    = ws + 1048576;       // 524288
    float* vv    = ws + 1572864;       // 524288
    float* skip  = ws + 2097152;       // 524288
    float* xA    = ws + 2621440;       // 524288
    float* xB    = ws + 3145728;       // 524288

    k_prep<<<512, 256, 0, stream>>>(adj, Et, Atf);
    k_hnode<<<BB * NN, 128, 0, stream>>>(node_obs, entity_embed, emb_W1, emb_b1, hnode);
    k_emb<<<BB * NN, 256, 0, stream>>>(hnode, Et, Atf, emb_W1, ln1_g, ln1_b,
                                       emb_Wh, emb_bh, lnh_g, lnh_b, x0);

    const float* xin = x0;
    int Kdim = EMB_H, ldx = EMB_H;
    float* bufs[3] = {xA, xB, xA};
    for (int l = 0; l < 3; ++l) {
        const float* Wq  = (const float*)d_in[12 + 9 * l + 0];
        const float* bq  = (const float*)d_in[12 + 9 * l + 1];
        const float* Wk  = (const float*)d_in[12 + 9 * l + 2];
        const float* bk  = (const float*)d_in[12 + 9 * l + 3];
        const float* Wv  = (const float*)d_in[12 + 9 * l + 4];
        const float* bv  = (const float*)d_in[12 + 9 * l + 5];
        const float* We  = (const float*)d_in[12 + 9 * l + 6];
        const float* Wsk = (const float*)d_in[12 + 9 * l + 7];
        const float* bsk = (const float*)d_in[12 + 9 * l + 8];
        dim3 g(8, 8);
        k_gemm_bias<<<g, 256, 0, stream>>>(xin, ldx, Wq,  bq,  q,    Kdim);
        k_gemm_bias<<<g, 256, 0, stream>>>(xin, ldx, Wk,  bk,  kk,   Kdim);
        k_gemm_bias<<<g, 256, 0, stream>>>(xin, ldx, Wv,  bv,  vv,   Kdim);
        k_gemm_bias<<<g, 256, 0, stream>>>(xin, ldx, Wsk, bsk, skip, Kdim);
        float* xout = bufs[l];
        k_attn<<<dim3(4, HEADS, BB), 256, 0, stream>>>(q, kk, vv, skip, Et, Atf, We, xout);
        xin = xout; Kdim = CCH; ldx = CCH;
    }
    k_gather<<<(BB * CCH + 255) / 256, 256, 0, stream>>>(xin, agent, (float*)d_out);
}